// AdditiveAttention_10660108829403
// MI455X (gfx1250) — compile-verified
//
#include <hip/hip_runtime.h>

typedef float v2f __attribute__((ext_vector_type(2)));
typedef float v8f __attribute__((ext_vector_type(8)));

#define NEG_INF_F (-1e30f)

// tanh via the native transcendental pipe:
// tanh(|x|) = (e-1)/(e+1) with e = exp2(2*|x|*log2(e)); sign restored.
// Saturates to +/-1 for large |x| (e -> inf => t -> 1).
__device__ __forceinline__ float fast_tanh(float x) {
    float ax = __builtin_fabsf(x);
    float e  = __builtin_amdgcn_exp2f(ax * 2.8853900817779268f); // 2/ln(2)
    float t  = 1.0f - 2.0f * __builtin_amdgcn_rcpf(e + 1.0f);
    return __builtin_copysignf(t, x);
}

// ---------------------------------------------------------------------------
// Generic f32 WMMA GEMM: C[M,N] = A[M,K] * B[K,N], one 16x16 tile per wave.
// Fragment layouts per CDNA5 ISA 7.12.2 (16x4 f32 A, 4x16 B, 16x16 f32 C/D):
//   A: lane l holds A[m0 + l%16][kk + 2*(l/16) + j] in vgpr j (j=0,1)
//   B: lane l holds B[kk + 2*(l/16) + j][n0 + l%16] in vgpr j
//   D: vgpr i, lane l -> C[m0 + i + 8*(l/16)][n0 + l%16]
// ---------------------------------------------------------------------------
__global__ __launch_bounds__(256) void wmma_gemm_f32(
    const float* __restrict__ A, const float* __restrict__ B,
    float* __restrict__ C,
    int M, int N, int Kdim, int lda, int ldb, int ldc,
    long long sA, long long sB, long long sC)
{
    A += (long long)blockIdx.z * sA;
    B += (long long)blockIdx.z * sB;
    C += (long long)blockIdx.z * sC;

    const int wave = threadIdx.x >> 5;
    const int lane = threadIdx.x & 31;
    const int ntn  = N >> 4;
    const int tile = blockIdx.x * 8 + wave;
    if (tile >= (M >> 4) * ntn) return;
    const int tm = tile / ntn;
    const int tn = tile - tm * ntn;
    const int m0 = tm << 4, n0 = tn << 4;
    const int r  = lane & 15;   // m (for A) / n (for B/C)
    const int h  = lane >> 4;   // selects K pair 0..1 vs 2..3

    const float* arow = A + (long long)(m0 + r) * lda + 2 * h;
    const float* bcol = B + (long long)(2 * h) * ldb + (n0 + r);

    v8f acc = {};
    #pragma unroll 4
    for (int kk = 0; kk < Kdim; kk += 4) {
        v2f a, b;
        a.x = arow[kk];                         // contiguous pair -> b64 load
        a.y = arow[kk + 1];
        b.x = bcol[(long long)kk * ldb];
        b.y = bcol[(long long)kk * ldb + ldb];
        acc = __builtin_amdgcn_wmma_f32_16x16x4_f32(
            /*neg_a=*/false, a, /*neg_b=*/false, b,
            /*c_mod=*/(short)0, acc, /*reuse_a=*/false, /*reuse_b=*/false);
    }

    float* crow = C + (long long)(m0 + h * 8) * ldc + (n0 + r);
    #pragma unroll
    for (int i = 0; i < 8; ++i)
        crow[(long long)i * ldc] = acc[i];
}

// ---------------------------------------------------------------------------
// Fused energy + mask + softmax.  One block per (b,q) row; 256 threads,
// each owning 4 keys.  q_proj row and v are staged in LDS (broadcast reads);
// k_proj rows stream from L2 (4 MB total -> fully L2 resident).
// ---------------------------------------------------------------------------
__global__ __launch_bounds__(256) void energy_softmax_kernel(
    const float* __restrict__ qproj,   // [B*Lq, 512]
    const float* __restrict__ kproj,   // [B*Lk, 512]
    const int*   __restrict__ mask,    // [B, 1024]
    const float* __restrict__ vvec,    // [512]
    float*       __restrict__ attn)    // [B*Lq, 1024]
{
    __shared__ float qs[512];
    __shared__ float vs[512];
    __shared__ float red[8];

    const int row  = blockIdx.x;     // b*256 + q
    const int b    = row >> 8;
    const int t    = threadIdx.x;
    const int lane = t & 31;
    const int wid  = t >> 5;

    qs[t]       = qproj[(long long)row * 512 + t];
    qs[t + 256] = qproj[(long long)row * 512 + t + 256];
    vs[t]       = vvec[t];
    vs[t + 256] = vvec[t + 256];
    __syncthreads();

    const int* mrow = mask + b * 1024;
    float e[4];
    #pragma unroll
    for (int kt = 0; kt < 4; ++kt) {
        const int k = kt * 256 + t;
        const float* krow = kproj + (long long)(b * 1024 + k) * 512;
        float acc = 0.0f;
        for (int f = 0; f < 512; f += 4) {
            float4 kv = *(const float4*)(krow + f);
            acc += vs[f + 0] * fast_tanh(qs[f + 0] + kv.x);
            acc += vs[f + 1] * fast_tanh(qs[f + 1] + kv.y);
            acc += vs[f + 2] * fast_tanh(qs[f + 2] + kv.z);
            acc += vs[f + 3] * fast_tanh(qs[f + 3] + kv.w);
        }
        e[kt] = (mrow[k] == 0) ? NEG_INF_F : acc;
    }

    // row max: wave32 shuffle tree, then cross-wave via LDS
    float m = fmaxf(fmaxf(e[0], e[1]), fmaxf(e[2], e[3]));
    #pragma unroll
    for (int off = 16; off > 0; off >>= 1)
        m = fmaxf(m, __shfl_xor(m, off, 32));
    if (lane == 0) red[wid] = m;
    __syncthreads();
    float rowmax = red[0];
    #pragma unroll
    for (int i = 1; i < 8; ++i) rowmax = fmaxf(rowmax, red[i]);
    __syncthreads();

    float p[4];
    float s = 0.0f;
    #pragma unroll
    for (int kt = 0; kt < 4; ++kt) {
        p[kt] = __builtin_amdgcn_exp2f((e[kt] - rowmax) * 1.4426950408889634f);
        s += p[kt];
    }
    #pragma unroll
    for (int off = 16; off > 0; off >>= 1)
        s += __shfl_xor(s, off, 32);
    if (lane == 0) red[wid] = s;
    __syncthreads();
    float rowsum = red[0];
    #pragma unroll
    for (int i = 1; i < 8; ++i) rowsum += red[i];

    float inv = __builtin_amdgcn_rcpf(rowsum);
    inv = inv * (2.0f - rowsum * inv);   // one Newton step

    float* arow = attn + (long long)row * 1024;
    #pragma unroll
    for (int kt = 0; kt < 4; ++kt)
        arow[kt * 256 + t] = p[kt] * inv;
}

extern "C" void kernel_launch(void* const* d_in, const int* in_sizes, int n_in,
                              void* d_out, int out_size, void* d_ws, size_t ws_size,
                              hipStream_t stream)
{
    const float* Q   = (const float*)d_in[0];   // [2,256,1024]
    const float* K   = (const float*)d_in[1];   // [2,1024,1024]
    const float* V   = (const float*)d_in[2];   // [2,1024,1024]
    const int*   msk = (const int*)  d_in[3];   // [2,1024]
    const float* Wq  = (const float*)d_in[4];   // [1024,512]
    const float* Wk  = (const float*)d_in[5];   // [1024,512]
    const float* v   = (const float*)d_in[6];   // [512]

    float* context = (float*)d_out;                       // [2,256,1024]
    float* attn    = (float*)d_out + 2 * 256 * 1024;      // [2,256,1024]

    float* qproj = (float*)d_ws;                          // [512, 512]   (1 MB)
    float* kproj = qproj + 512 * 512;                     // [2048, 512]  (4 MB)

    // q_proj = Q @ Wq : (B*Lq=512) x 1024 x 512 -> 1024 tiles / 8 waves
    wmma_gemm_f32<<<dim3(128, 1, 1), 256, 0, stream>>>(
        Q, Wq, qproj, 512, 512, 1024, 1024, 512, 512, 0, 0, 0);

    // k_proj = K @ Wk : (B*Lk=2048) x 1024 x 512 -> 4096 tiles
    wmma_gemm_f32<<<dim3(512, 1, 1), 256, 0, stream>>>(
        K, Wk, kproj, 2048, 512, 1024, 1024, 512, 512, 0, 0, 0);

    // energy -> mask -> softmax, written in place into attn half of d_out
    energy_softmax_kernel<<<dim3(512, 1, 1), 256, 0, stream>>>(
        qproj, kproj, msk, v, attn);

    // context[b] = attn[b] @ V[b] : 256 x 1024 x 1024, batched over grid.z
    wmma_gemm_f32<<<dim3(128, 1, 2), 256, 0, stream>>>(
        attn, V, context, 256, 1024, 1024, 1024, 1024, 1024,
        (long long)256 * 1024, (long long)1024 * 1024, (long long)256 * 1024);
}